// Model_44367012167732
// MI455X (gfx1250) — compile-verified
//
#include <hip/hip_runtime.h>
#include <hip/hip_bf16.h>

typedef float v2f __attribute__((ext_vector_type(2)));
typedef float v8f __attribute__((ext_vector_type(8)));

#define B_DIM 64
#define S_DIM 256
#define T_DIM 8
#define D_DIM 256
#define M_DIM 2048
#define ROW_STRIDE (T_DIM * D_DIM)   // stride between type rows in token_reprs
#define MAX_MTILES 192               // worst case: 128 full tiles + 64 bin-pad tiles (<=188)
#define NEG_INF_F (-1.0e10f)

// ---------------------------------------------------------------------------
// K0: fold w1 @ w2 into a single D-vector v, and c = b1.w2 + b2 (no
// nonlinearity between the two linear layers => exact algebraic fusion).
// ---------------------------------------------------------------------------
__global__ void fold_weights(const float* __restrict__ w1, const float* __restrict__ b1,
                             const float* __restrict__ w2, const float* __restrict__ b2,
                             float* __restrict__ v /* [256] then c at [256] */) {
    int d = threadIdx.x;  // 256 threads
    float acc = 0.0f;
#pragma unroll 8
    for (int j = 0; j < 128; ++j)
        acc = fmaf(w1[d * 128 + j], w2[j], acc);
    v[d] = acc;
    if (d == 0) {
        float cacc = b2[0];
        for (int j = 0; j < 128; ++j)
            cacc = fmaf(b1[j], w2[j], cacc);
        v[D_DIM] = cacc;
    }
}

// ---------------------------------------------------------------------------
// K1: bin the M=2048 lm rows by batch pointer (64 bins), pad each bin to a
// multiple of 16 (-1 entries), emit per-16-row-tile batch id. Single block.
// ---------------------------------------------------------------------------
__global__ void bin_kernel(const int* __restrict__ bp,
                           int* __restrict__ order,    // [MAX_MTILES*16], -1 padded
                           int* __restrict__ tile_b,   // [MAX_MTILES]
                           int* __restrict__ n_mtiles) {
    __shared__ int cnt[B_DIM];
    __shared__ int cur[B_DIM];
    const int tid = threadIdx.x;  // 1024
    if (tid < B_DIM) cnt[tid] = 0;
    __syncthreads();
    for (int m = tid; m < M_DIM; m += 1024)
        atomicAdd(&cnt[bp[m]], 1);
    __syncthreads();
    if (tid == 0) {
        int pos = 0, tile = 0;
        for (int b = 0; b < B_DIM; ++b) {
            cur[b] = pos;
            int nt = (cnt[b] + 15) >> 4;
            for (int t = 0; t < nt; ++t) tile_b[tile++] = b;
            pos += nt << 4;
        }
        *n_mtiles = tile;
    }
    __syncthreads();
    for (int i = tid; i < MAX_MTILES * 16; i += 1024)
        order[i] = -1;
    __syncthreads();
    for (int m = tid; m < M_DIM; m += 1024) {
        int pos = atomicAdd(&cur[bp[m]], 1);
        order[pos] = m;  // each m placed exactly once -> deterministic outputs
    }
}

// ---------------------------------------------------------------------------
// K2: lemmas[e] = sum_d x[scope,d]*y[goal,d]*v[d] + c.  One wave32 per edge:
// each lane reads a contiguous 32B slice of each 1KB row (L2-resident table),
// then 5-step shuffle reduction.
// ---------------------------------------------------------------------------
__global__ void lemmas_kernel(const float* __restrict__ token, const int* __restrict__ edge,
                              const float* __restrict__ vws, float* __restrict__ out, int E) {
    __shared__ float vsh[D_DIM];
    __shared__ float csh;
    const int tid = threadIdx.x;  // 256 = 8 waves
    if (tid < D_DIM) vsh[tid] = vws[tid];
    if (tid == 0) csh = vws[D_DIM];
    __syncthreads();
    const int lane = tid & 31;
    const int e = blockIdx.x * 8 + (tid >> 5);
    if (e >= E) return;  // wave-uniform
    const float* __restrict__ x = token + (size_t)edge[e] * ROW_STRIDE;
    const float* __restrict__ y = token + (size_t)edge[E + e] * ROW_STRIDE;
    const int base = lane * 8;
    float4 x0 = *(const float4*)(x + base);
    float4 x1 = *(const float4*)(x + base + 4);
    float4 y0 = *(const float4*)(y + base);
    float4 y1 = *(const float4*)(y + base + 4);
    float4 v0 = *(const float4*)(vsh + base);
    float4 v1 = *(const float4*)(vsh + base + 4);
    float p = 0.0f;
    p = fmaf(x0.x * y0.x, v0.x, p);
    p = fmaf(x0.y * y0.y, v0.y, p);
    p = fmaf(x0.z * y0.z, v0.z, p);
    p = fmaf(x0.w * y0.w, v0.w, p);
    p = fmaf(x1.x * y1.x, v1.x, p);
    p = fmaf(x1.y * y1.y, v1.y, p);
    p = fmaf(x1.z * y1.z, v1.z, p);
    p = fmaf(x1.w * y1.w, v1.w, p);
#pragma unroll
    for (int off = 16; off > 0; off >>= 1)
        p += __shfl_xor(p, off, 32);
    if (lane == 0) out[e] = p + csh;
}

// ---------------------------------------------------------------------------
// K3: lm_preds via fp32 WMMA, 4 s-tiles per wave.  One block (4 waves) per
// m-tile; wave w owns s-columns [64w, 64w+64).  A (16 gathered mask rows) is
// loaded once and fed to 4 independent WMMA accumulator chains -> 5 loads per
// 4 WMMAs and no single serial D->C dependency chain.
// V_WMMA_F32_16X16X4_F32 operand layout:
//   A (16x4): lane<16 -> A[M=lane][k0,k0+1]; lane>=16 -> A[M=lane-16][k0+2,k0+3]
//   B (4x16): lane<16 -> B[0,1][N=lane];     lane>=16 -> B[2,3][N=lane-16]
//   D (16x16): acc[j] = D[M=j+8*half][N=lane&15]
// ---------------------------------------------------------------------------
__global__ void lmpred_kernel(const float* __restrict__ token, const int* __restrict__ lm_indices,
                              const unsigned char* __restrict__ pad,
                              const int* __restrict__ order, const int* __restrict__ tile_b,
                              const int* __restrict__ n_mtiles, float* __restrict__ out) {
    const int mtile = blockIdx.x;
    if (mtile >= *n_mtiles) return;  // block-uniform

    __shared__ int ent[16];
    __shared__ int arowidx[16];
    const int tid = threadIdx.x;  // 128 = 4 waves
    if (tid < 16) {
        int m = order[mtile * 16 + tid];
        ent[tid] = m;
        arowidx[tid] = (m >= 0) ? lm_indices[m] : 0;  // padded rows read row 0, never written
    }
    __syncthreads();

    const int wave = tid >> 5;    // s-tile group: columns [64*wave, 64*wave+64)
    const int lane = tid & 31;
    const int half = lane >> 4;   // 0: K pair (0,1); 1: K pair (2,3)
    const int l16 = lane & 15;
    const int b = tile_b[mtile];

    const float* __restrict__ arow = token + (size_t)arowidx[l16] * D_DIM + 2 * half;
    const int s0 = wave * 64 + l16;
    const float* __restrict__ brow0 = token + (size_t)(b * S_DIM + s0) * ROW_STRIDE + 2 * half;
    const float* __restrict__ brow1 = brow0 + (size_t)16 * ROW_STRIDE;
    const float* __restrict__ brow2 = brow0 + (size_t)32 * ROW_STRIDE;
    const float* __restrict__ brow3 = brow0 + (size_t)48 * ROW_STRIDE;

    v8f acc[4] = {v8f{}, v8f{}, v8f{}, v8f{}};
#pragma unroll 2
    for (int k0 = 0; k0 < D_DIM; k0 += 4) {
        v2f a  = *(const v2f*)(arow + k0);
        v2f b0 = *(const v2f*)(brow0 + k0);
        v2f b1 = *(const v2f*)(brow1 + k0);
        v2f b2 = *(const v2f*)(brow2 + k0);
        v2f b3 = *(const v2f*)(brow3 + k0);
        acc[0] = __builtin_amdgcn_wmma_f32_16x16x4_f32(false, a, false, b0, (short)0, acc[0], false, false);
        acc[1] = __builtin_amdgcn_wmma_f32_16x16x4_f32(false, a, false, b1, (short)0, acc[1], false, false);
        acc[2] = __builtin_amdgcn_wmma_f32_16x16x4_f32(false, a, false, b2, (short)0, acc[2], false, false);
        acc[3] = __builtin_amdgcn_wmma_f32_16x16x4_f32(false, a, false, b3, (short)0, acc[3], false, false);
    }

#pragma unroll
    for (int q = 0; q < 4; ++q) {
        const int s = wave * 64 + q * 16 + l16;
        const bool ok = pad[b * S_DIM + s] != 0;
#pragma unroll
        for (int j = 0; j < 8; ++j) {
            int m = ent[j + 8 * half];
            if (m >= 0)
                out[(size_t)m * S_DIM + s] = ok ? acc[q][j] : NEG_INF_F;
        }
    }
}

// ---------------------------------------------------------------------------
extern "C" void kernel_launch(void* const* d_in, const int* in_sizes, int n_in,
                              void* d_out, int out_size, void* d_ws, size_t ws_size,
                              hipStream_t stream) {
    const float* token       = (const float*)d_in[0];  // (B,S,T,D) f32
    const float* w1          = (const float*)d_in[1];  // (256,128)
    const float* b1          = (const float*)d_in[2];  // (128,)
    const float* w2          = (const float*)d_in[3];  // (128,1)
    const float* b2          = (const float*)d_in[4];  // (1,)
    const int*   edge        = (const int*)d_in[5];    // (2,E)
    const int*   lm_indices  = (const int*)d_in[6];    // (M,)
    const int*   bp          = (const int*)d_in[7];    // (M,)
    const unsigned char* pad = (const unsigned char*)d_in[8];  // (B,S) bool

    const int E = in_sizes[5] / 2;

    // workspace layout
    float* v       = (float*)d_ws;                       // 256 + 1 (c)
    int*   ws_i    = (int*)((char*)d_ws + (D_DIM + 1) * sizeof(float));
    int* n_mtiles  = ws_i;                               // 1
    int* tile_b    = ws_i + 1;                           // MAX_MTILES
    int* order     = ws_i + 1 + MAX_MTILES;              // MAX_MTILES*16

    float* lemmas   = (float*)d_out;        // [E]
    float* lm_preds = lemmas + E;           // [M*S]

    fold_weights<<<1, 256, 0, stream>>>(w1, b1, w2, b2, v);
    bin_kernel<<<1, 1024, 0, stream>>>(bp, order, tile_b, n_mtiles);
    lemmas_kernel<<<(E + 7) / 8, 256, 0, stream>>>(token, edge, v, lemmas, E);
    lmpred_kernel<<<MAX_MTILES, 128, 0, stream>>>(token, lm_indices, pad, order,
                                                  tile_b, n_mtiles, lm_preds);
}